// PKM_5574867550364
// MI455X (gfx1250) — compile-verified
//
#include <hip/hip_runtime.h>
#include <math.h>

typedef __attribute__((ext_vector_type(2))) float v2f;
typedef __attribute__((ext_vector_type(8))) float v8f;

#define PKM_DIM   512
#define PKM_HEADS 4
#define PKM_NK    256
#define PKM_TK    32
#define PKM_DH    128
#define PKM_BT    4096
#define PKM_DQ    1024

static __device__ __forceinline__ v8f wmma4(v2f a, v2f b, v8f c) {
  // V_WMMA_F32_16X16X4_F32 : D = A(16x4,f32) * B(4x16,f32) + C(16x16,f32)
  return __builtin_amdgcn_wmma_f32_16x16x4_f32(false, a, false, b, (short)0, c,
                                               false, false);
}

// ---------------------------------------------------------------------------
// Kernel 1: Q = X(4096x512) @ WQ^T(512x1024).  One wave computes a 16x64
// strip: 4 accumulators share one A fragment per K-step.
// ---------------------------------------------------------------------------
__global__ void pkm_qproj(const float* __restrict__ X,
                          const float* __restrict__ WQ,
                          float* __restrict__ Q) {
  const int wave = (blockIdx.x * blockDim.x + threadIdx.x) >> 5;
  const int lane = threadIdx.x & 31;
  const int mt = wave >> 4;            // 256 M tiles
  const int ng = wave & 15;            // 16 N groups of 64
  const int m0 = mt << 4, n0 = ng << 6;
  const int hi = lane >> 4, lo = lane & 15;

  v8f a0 = {}, a1 = {}, a2 = {}, a3 = {};
  const float* ap = X + (size_t)(m0 + lo) * PKM_DIM + (hi << 1);
  const float* bp = WQ + (size_t)(n0 + lo) * PKM_DIM + (hi << 1);
#pragma unroll 4
  for (int k = 0; k < PKM_DIM; k += 4) {
    v2f av = *(const v2f*)(ap + k);
    v2f b0 = *(const v2f*)(bp + k);
    v2f b1 = *(const v2f*)(bp + 16 * PKM_DIM + k);
    v2f b2 = *(const v2f*)(bp + 32 * PKM_DIM + k);
    v2f b3 = *(const v2f*)(bp + 48 * PKM_DIM + k);
    a0 = wmma4(av, b0, a0);
    a1 = wmma4(av, b1, a1);
    a2 = wmma4(av, b2, a2);
    a3 = wmma4(av, b3, a3);
  }
#pragma unroll
  for (int r = 0; r < 8; ++r) {
    float* q = Q + (size_t)(m0 + r + (hi << 3)) * PKM_DQ + n0 + lo;
    q[0] = a0[r];
    q[16] = a1[r];
    q[32] = a2[r];
    q[48] = a3[r];
  }
}

// ---------------------------------------------------------------------------
// Kernel 2: LayerNorm each 128-vector of Q in place.  One wave per vector.
// ---------------------------------------------------------------------------
__global__ void pkm_ln(float* __restrict__ Q, const float* __restrict__ g,
                       const float* __restrict__ b) {
  const int wave = (blockIdx.x * blockDim.x + threadIdx.x) >> 5;
  const int lane = threadIdx.x & 31;
  float4* v = (float4*)(Q + (size_t)wave * PKM_DH);
  float4 x = v[lane];
  float s = x.x + x.y + x.z + x.w;
  float s2 = x.x * x.x + x.y * x.y + x.z * x.z + x.w * x.w;
#pragma unroll
  for (int off = 16; off > 0; off >>= 1) {
    s += __shfl_xor(s, off, 32);
    s2 += __shfl_xor(s2, off, 32);
  }
  const float mu = s * (1.0f / PKM_DH);
  const float var = s2 * (1.0f / PKM_DH) - mu * mu;
  const float rs = rsqrtf(var + 1e-5f);
  const int d = lane << 2;
  x.x = (x.x - mu) * rs * g[d + 0] + b[d + 0];
  x.y = (x.y - mu) * rs * g[d + 1] + b[d + 1];
  x.z = (x.z - mu) * rs * g[d + 2] + b[d + 2];
  x.w = (x.w - mu) * rs * g[d + 3] + b[d + 3];
  v[lane] = x;
}

// ---------------------------------------------------------------------------
// Kernel 3: dots[bt,h,p,n] = sum_d Q[bt, p*512+h*128+d] * keys[h,n,p,d].
// Per (h,p): (4096x128)@(128x256).  Same 16x64-per-wave WMMA scheme.
// ---------------------------------------------------------------------------
__global__ void pkm_dots(const float* __restrict__ Q,
                         const float* __restrict__ keys,
                         float* __restrict__ dots) {
  const int wave = (blockIdx.x * blockDim.x + threadIdx.x) >> 5;
  const int lane = threadIdx.x & 31;
  const int mt = wave >> 5;            // 256 M tiles
  const int rem = wave & 31;
  const int hp = rem >> 2;             // h*2+p
  const int ng = rem & 3;              // 4 N groups of 64
  const int h = hp >> 1, p = hp & 1;
  const int m0 = mt << 4, n0 = ng << 6;
  const int hi = lane >> 4, lo = lane & 15;
  const int bstride = 2 * PKM_DH;      // stride between key rows (floats)

  v8f a0 = {}, a1 = {}, a2 = {}, a3 = {};
  const float* ap =
      Q + (size_t)(m0 + lo) * PKM_DQ + p * 512 + h * 128 + (hi << 1);
  const float* bp =
      keys + ((size_t)(h * PKM_NK + n0 + lo) * 2 + p) * PKM_DH + (hi << 1);
#pragma unroll 4
  for (int k = 0; k < PKM_DH; k += 4) {
    v2f av = *(const v2f*)(ap + k);
    v2f b0 = *(const v2f*)(bp + k);
    v2f b1 = *(const v2f*)(bp + 16 * bstride + k);
    v2f b2 = *(const v2f*)(bp + 32 * bstride + k);
    v2f b3 = *(const v2f*)(bp + 48 * bstride + k);
    a0 = wmma4(av, b0, a0);
    a1 = wmma4(av, b1, a1);
    a2 = wmma4(av, b2, a2);
    a3 = wmma4(av, b3, a3);
  }
#pragma unroll
  for (int r = 0; r < 8; ++r) {
    float* dp = dots +
                ((size_t)((m0 + r + (hi << 3)) * PKM_HEADS + h) * 2 + p) *
                    PKM_NK +
                n0 + lo;
    dp[0] = a0[r];
    dp[16] = a1[r];
    dp[32] = a2[r];
    dp[48] = a3[r];
  }
}

// ---------------------------------------------------------------------------
// Kernel 4: two-stage top-k + softmax.  One wave per (bt,h).
// Iterative selection: lane k keeps the k-th largest.  Ties -> lower index
// (matches lax.top_k).  All state in registers (masks are unrolled selects).
// ---------------------------------------------------------------------------
__global__ void pkm_topk(const float* __restrict__ dots,
                         float* __restrict__ W, int* __restrict__ IDX) {
  const int wave = (blockIdx.x * blockDim.x + threadIdx.x) >> 5;  // bt*4+h
  const int lane = threadIdx.x & 31;
  const float* base = dots + (size_t)wave * (2 * PKM_NK);

  float r0[8], r1[8];
#pragma unroll
  for (int j = 0; j < 8; ++j) {
    r0[j] = base[lane + (j << 5)];
    r1[j] = base[PKM_NK + lane + (j << 5)];
  }

  float sx = 0.f, sy = 0.f;
  int ix = 0, iy = 0;

  // top-32 of p=0 scores
#pragma unroll 1
  for (int k = 0; k < PKM_TK; ++k) {
    float m = r0[0];
    int a = 0;
#pragma unroll
    for (int j = 1; j < 8; ++j)
      if (r0[j] > m) { m = r0[j]; a = j; }
    int gdx = lane + (a << 5);
#pragma unroll
    for (int off = 16; off > 0; off >>= 1) {
      float om = __shfl_xor(m, off, 32);
      int og = __shfl_xor(gdx, off, 32);
      if (om > m || (om == m && og < gdx)) { m = om; gdx = og; }
    }
    if (lane == k) { sx = m; ix = gdx; }
#pragma unroll
    for (int j = 0; j < 8; ++j)
      if (gdx == lane + (j << 5)) r0[j] = -INFINITY;
  }

  // top-32 of p=1 scores
#pragma unroll 1
  for (int k = 0; k < PKM_TK; ++k) {
    float m = r1[0];
    int a = 0;
#pragma unroll
    for (int j = 1; j < 8; ++j)
      if (r1[j] > m) { m = r1[j]; a = j; }
    int gdx = lane + (a << 5);
#pragma unroll
    for (int off = 16; off > 0; off >>= 1) {
      float om = __shfl_xor(m, off, 32);
      int og = __shfl_xor(gdx, off, 32);
      if (om > m || (om == m && og < gdx)) { m = om; gdx = og; }
    }
    if (lane == k) { sy = m; iy = gdx; }
#pragma unroll
    for (int j = 0; j < 8; ++j)
      if (gdx == lane + (j << 5)) r1[j] = -INFINITY;
  }

  // outer sum: lane j holds column j -> cand[i] = sx[i] + sy[j]
  float cand[32];
#pragma unroll
  for (int i = 0; i < 32; ++i) cand[i] = __shfl(sx, i, 32) + sy;

  // top-32 of the 1024 combined scores (flat index f = i*32 + j)
  float fs = 0.f;
  int fpos = 0;
#pragma unroll 1
  for (int k = 0; k < PKM_TK; ++k) {
    float m = cand[0];
    int a = 0;
#pragma unroll
    for (int i = 1; i < 32; ++i)
      if (cand[i] > m) { m = cand[i]; a = i; }
    int f = (a << 5) + lane;
#pragma unroll
    for (int off = 16; off > 0; off >>= 1) {
      float om = __shfl_xor(m, off, 32);
      int of = __shfl_xor(f, off, 32);
      if (om > m || (om == m && of < f)) { m = om; f = of; }
    }
    if (lane == k) { fs = m; fpos = f; }
#pragma unroll
    for (int i = 0; i < 32; ++i)
      if (f == (i << 5) + lane) cand[i] = -INFINITY;
  }

  // combined value index: ix from lane fpos>>5, iy from lane fpos&31
  const int ii = __shfl(ix, fpos >> 5, 32);
  const int jj = __shfl(iy, fpos & 31, 32);
  const int vidx = ii * PKM_NK + jj;

  // softmax (fs sorted descending -> max at lane 0)
  const float mx = __shfl(fs, 0, 32);
  const float e = __expf(fs - mx);
  float sum = e;
#pragma unroll
  for (int off = 16; off > 0; off >>= 1) sum += __shfl_xor(sum, off, 32);

  W[(size_t)wave * PKM_TK + lane] = e / sum;
  IDX[(size_t)wave * PKM_TK + lane] = vidx;
}

// ---------------------------------------------------------------------------
// Kernel 5: out[bt,:] = sum over 128 (h,k) of w * values[idx,:].
// One 128-thread block per token; each thread owns one float4 column slice;
// every gathered row is a fully-coalesced 2KB burst (table is L2-resident).
// ---------------------------------------------------------------------------
__global__ void pkm_gather(const float* __restrict__ values,
                           const float* __restrict__ W,
                           const int* __restrict__ IDX,
                           float* __restrict__ out) {
  __shared__ float sw[128];
  __shared__ int si[128];
  const int bt = blockIdx.x;
  const int t = threadIdx.x;  // 0..127
  sw[t] = W[(size_t)bt * 128 + t];
  si[t] = IDX[(size_t)bt * 128 + t];
  __syncthreads();

  float4 acc = {0.f, 0.f, 0.f, 0.f};
  const float4* v4 = (const float4*)values;
#pragma unroll 4
  for (int e = 0; e < 128; ++e) {
    const float w = sw[e];
    const float4 v = v4[(size_t)si[e] * 128 + t];
    acc.x += w * v.x;
    acc.y += w * v.y;
    acc.z += w * v.z;
    acc.w += w * v.w;
  }
  ((float4*)out)[(size_t)bt * 128 + t] = acc;
}

// ---------------------------------------------------------------------------
extern "C" void kernel_launch(void* const* d_in, const int* in_sizes, int n_in,
                              void* d_out, int out_size, void* d_ws,
                              size_t ws_size, hipStream_t stream) {
  (void)in_sizes; (void)n_in; (void)out_size; (void)ws_size;
  const float* x = (const float*)d_in[0];
  const float* wq = (const float*)d_in[1];
  const float* ln_g = (const float*)d_in[2];
  const float* ln_b = (const float*)d_in[3];
  const float* keys = (const float*)d_in[4];
  const float* values = (const float*)d_in[5];
  float* out = (float*)d_out;

  char* ws = (char*)d_ws;
  const size_t q_bytes = (size_t)PKM_BT * PKM_DQ * sizeof(float);        // 16 MB
  const size_t d_bytes = (size_t)PKM_BT * 8 * PKM_NK * sizeof(float);    // 32 MB
  const size_t w_bytes = (size_t)PKM_BT * PKM_HEADS * PKM_TK * sizeof(float);
  float* Q = (float*)ws;
  float* dots = (float*)(ws + q_bytes);
  float* W = (float*)(ws + q_bytes + d_bytes);
  int* IDX = (int*)(ws + q_bytes + d_bytes + w_bytes);

  // 1) Q projection: 4096 waves (256 Mtiles x 16 Ngroups) -> 512 blocks x 8w
  pkm_qproj<<<512, 256, 0, stream>>>(x, wq, Q);
  // 2) LayerNorm: 32768 vectors -> 4096 blocks x 8 waves
  pkm_ln<<<4096, 256, 0, stream>>>(Q, ln_g, ln_b);
  // 3) dots: 8192 waves (256 Mtiles x 8 hp x 4 Ngroups) -> 1024 blocks
  pkm_dots<<<1024, 256, 0, stream>>>(Q, keys, dots);
  // 4) top-k + softmax: 16384 waves -> 2048 blocks
  pkm_topk<<<2048, 256, 0, stream>>>(dots, W, IDX);
  // 5) gather: one block per token
  pkm_gather<<<PKM_BT, 128, 0, stream>>>(values, W, IDX, out);
}